// AttentionBlock_25391846654757
// MI455X (gfx1250) — compile-verified
//
#include <hip/hip_runtime.h>

#define B_ 2
#define C_ 256
#define N_ 4096
#define HEADS_ 4
#define HD_ 64
#define GROUPS_ 8
#define CPG_ 32
#define EPS_ 1e-5f

typedef __attribute__((ext_vector_type(2))) float v2f;
typedef __attribute__((ext_vector_type(8))) float v8f;

__device__ __forceinline__ v8f wmma_f32(v2f a, v2f b, v8f c) {
  // D = A(16x4) * B(4x16) + C(16x16), full fp32
  return __builtin_amdgcn_wmma_f32_16x16x4_f32(
      /*neg_a=*/false, a, /*neg_b=*/false, b,
      /*c_mod=*/(short)0, c, /*reuse_a=*/false, /*reuse_b=*/false);
}

// ---------------------------------------------------------------------------
// Kernel 1: GroupNorm. One block per (batch, group). 32 ch x 4096 = 131072 el.
// ---------------------------------------------------------------------------
__global__ __launch_bounds__(256) void gn_kernel(
    const float* __restrict__ x, const float* __restrict__ gamma,
    const float* __restrict__ beta, float* __restrict__ xn) {
  __shared__ float s1[256], s2[256];
  const int b = blockIdx.x / GROUPS_;
  const int g = blockIdx.x % GROUPS_;
  const int count = CPG_ * N_;
  const size_t base = ((size_t)b * C_ + (size_t)g * CPG_) * N_;
  float sum = 0.f, ssq = 0.f;
  for (int i = threadIdx.x; i < count; i += 256) {
    float vv = x[base + i];
    sum += vv;
    ssq += vv * vv;
  }
  s1[threadIdx.x] = sum;
  s2[threadIdx.x] = ssq;
  __syncthreads();
  for (int off = 128; off > 0; off >>= 1) {
    if ((int)threadIdx.x < off) {
      s1[threadIdx.x] += s1[threadIdx.x + off];
      s2[threadIdx.x] += s2[threadIdx.x + off];
    }
    __syncthreads();
  }
  const float mean = s1[0] / (float)count;
  const float var  = s2[0] / (float)count - mean * mean;
  const float rstd = rsqrtf(var + EPS_);
  for (int i = threadIdx.x; i < count; i += 256) {
    int c = g * CPG_ + (i >> 12);  // i / N_
    float vv = (x[base + i] - mean) * rstd;
    xn[base + i] = vv * gamma[c] + beta[c];
  }
}

// ---------------------------------------------------------------------------
// Kernel 2: QKV GEMM. out[o,n] = sum_c w[o,c]*xn[b,c,n] + bias[o]
// M=768 (o), K=256 (c), N=4096 (n). 64x64 tile per WG, 16x64 strip per wave.
// Activation tile stored transposed in LDS so B-fragments are contiguous b64.
// Output scattered into per-head [b][h][n][d] q/k/v buffers.
// ---------------------------------------------------------------------------
__global__ __launch_bounds__(128) void qkv_kernel(
    const float* __restrict__ xn, const float* __restrict__ w,
    const float* __restrict__ bias, float* __restrict__ q,
    float* __restrict__ k, float* __restrict__ v) {
  __shared__ float Wl[64][68];  // [o][c]   (row-major, A frags contiguous)
  __shared__ float Xt[64][68];  // [n][c]   (transposed, B frags contiguous)
  const int n0 = blockIdx.x * 64;
  const int o0 = blockIdx.y * 64;
  const int b  = blockIdx.z;
  const int tid = threadIdx.x;
  const int wid = tid >> 5, lane = tid & 31, l16 = lane & 15, half = lane >> 4;
  const int orow = o0 + wid * 16;
  const float* xb = xn + (size_t)b * C_ * N_;

  v8f acc[4] = {};
  for (int kb = 0; kb < C_; kb += 64) {
    for (int i = tid * 4; i < 64 * 64; i += 128 * 4) {
      int r = i >> 6, cc = i & 63;
      *(float4*)&Wl[r][cc] = *(const float4*)(w + (size_t)(o0 + r) * C_ + kb + cc);
      float4 xv = *(const float4*)(xb + (size_t)(kb + r) * N_ + n0 + cc);
      Xt[cc + 0][r] = xv.x;
      Xt[cc + 1][r] = xv.y;
      Xt[cc + 2][r] = xv.z;
      Xt[cc + 3][r] = xv.w;
    }
    __syncthreads();
#pragma unroll
    for (int ks = 0; ks < 16; ++ks) {
      const int kd = 4 * ks + 2 * half;
      const v2f a = *(const v2f*)&Wl[wid * 16 + l16][kd];
#pragma unroll
      for (int nt = 0; nt < 4; ++nt) {
        const v2f bb = *(const v2f*)&Xt[nt * 16 + l16][kd];
        acc[nt] = wmma_f32(a, bb, acc[nt]);
      }
    }
    __syncthreads();
  }

  // Epilogue: add bias, scatter to q/k/v as [b][h][n][d]
  const int which = orow >> 8;          // 0=q, 1=k, 2=v
  const int hh    = (orow & 255) >> 6;  // head
  const int dbase = (orow & 63) + 8 * half;
  float* outb = (which == 0) ? q : (which == 1) ? k : v;
  float bv[8];
#pragma unroll
  for (int vv = 0; vv < 8; ++vv) bv[vv] = bias[orow + 8 * half + vv];
#pragma unroll
  for (int nt = 0; nt < 4; ++nt) {
    const int n = n0 + nt * 16 + l16;
    float* p = outb + (((size_t)b * HEADS_ + hh) * N_ + n) * HD_ + dbase;
    float4 lo, hi;
    lo.x = acc[nt][0] + bv[0]; lo.y = acc[nt][1] + bv[1];
    lo.z = acc[nt][2] + bv[2]; lo.w = acc[nt][3] + bv[3];
    hi.x = acc[nt][4] + bv[4]; hi.y = acc[nt][5] + bv[5];
    hi.z = acc[nt][6] + bv[6]; hi.w = acc[nt][7] + bv[7];
    *(float4*)p = lo;
    *(float4*)(p + 4) = hi;
  }
}

// ---------------------------------------------------------------------------
// Kernel 3: flash attention. One WG per (b, h, 64-query block); each wave
// owns a 16-query strip (softmax stats stay within 16-lane halves).
// K tile row-major (QK^T B-frags contiguous), V tile transposed (PV B-frags
// contiguous), P strip per-wave in LDS (PV A-frags contiguous).
// ---------------------------------------------------------------------------
__global__ __launch_bounds__(128) void attn_kernel(
    const float* __restrict__ q, const float* __restrict__ k,
    const float* __restrict__ v, float* __restrict__ ao) {
  __shared__ float Kl[64][68];      // [key][d]
  __shared__ float Vt[64][68];      // [d][key]  (transposed)
  __shared__ float Pl[4][16][68];   // per-wave P strip [qrow][key]
  const int q0 = blockIdx.x * 64;
  const int h  = blockIdx.y;
  const int b  = blockIdx.z;
  const int tid = threadIdx.x;
  const int wid = tid >> 5, lane = tid & 31, l16 = lane & 15, half = lane >> 4;
  const size_t hb = ((size_t)b * HEADS_ + h) * (size_t)N_ * HD_;
  const float* qb = q + hb;
  const float* kb = k + hb;
  const float* vb = v + hb;
  const float scale = 0.125f;  // hd^-0.5

  // Q strip as pre-scaled A fragments (16 k-steps of depth 4)
  v2f qa[16];
  const int qrow = q0 + wid * 16 + l16;
#pragma unroll
  for (int ks = 0; ks < 16; ++ks) {
    const int d = 4 * ks + 2 * half;
    const v2f t = *(const v2f*)(qb + (size_t)qrow * HD_ + d);
    qa[ks] = t * scale;
  }

  v8f oacc[4] = {};
  float rm[8], rs[8];
#pragma unroll
  for (int i = 0; i < 8; ++i) { rm[i] = -1e30f; rs[i] = 0.f; }

  for (int j = 0; j < N_; j += 64) {
    for (int i = tid * 4; i < 64 * 64; i += 512) {
      int r = i >> 6, cc = i & 63;
      *(float4*)&Kl[r][cc] = *(const float4*)(kb + (size_t)(j + r) * HD_ + cc);
      float4 vv4 = *(const float4*)(vb + (size_t)(j + r) * HD_ + cc);
      Vt[cc + 0][r] = vv4.x;
      Vt[cc + 1][r] = vv4.y;
      Vt[cc + 2][r] = vv4.z;
      Vt[cc + 3][r] = vv4.w;
    }
    __syncthreads();

    // S = Qs * K^T  (16 queries x 64 keys per wave)
    v8f S[4] = {};
#pragma unroll
    for (int ks = 0; ks < 16; ++ks) {
      const int kd = 4 * ks + 2 * half;
#pragma unroll
      for (int nt = 0; nt < 4; ++nt) {
        const v2f bb = *(const v2f*)&Kl[nt * 16 + l16][kd];
        S[nt] = wmma_f32(qa[ks], bb, S[nt]);
      }
    }

    // Online softmax. Element (v) in this lane belongs to row v + 8*half.
#pragma unroll
    for (int vv = 0; vv < 8; ++vv) {
      float cand = fmaxf(fmaxf(S[0][vv], S[1][vv]), fmaxf(S[2][vv], S[3][vv]));
      cand = fmaxf(cand, __shfl_xor(cand, 1, 32));
      cand = fmaxf(cand, __shfl_xor(cand, 2, 32));
      cand = fmaxf(cand, __shfl_xor(cand, 4, 32));
      cand = fmaxf(cand, __shfl_xor(cand, 8, 32));
      const float nm = fmaxf(rm[vv], cand);
      const float f  = __expf(rm[vv] - nm);
      rs[vv] *= f;
      float psum = 0.f;
#pragma unroll
      for (int nt = 0; nt < 4; ++nt) {
        float p = __expf(S[nt][vv] - nm);
        Pl[wid][vv + 8 * half][nt * 16 + l16] = p;
        psum += p;
      }
      psum += __shfl_xor(psum, 1, 32);
      psum += __shfl_xor(psum, 2, 32);
      psum += __shfl_xor(psum, 4, 32);
      psum += __shfl_xor(psum, 8, 32);
      rs[vv] += psum;
      rm[vv] = nm;
#pragma unroll
      for (int dt = 0; dt < 4; ++dt) oacc[dt][vv] *= f;
    }

    // O += P * V (wave-private Pl; no barrier needed within wave)
#pragma unroll
    for (int ks = 0; ks < 16; ++ks) {
      const int kk = 4 * ks + 2 * half;
      const v2f a = *(const v2f*)&Pl[wid][l16][kk];
#pragma unroll
      for (int dt = 0; dt < 4; ++dt) {
        const v2f bb = *(const v2f*)&Vt[dt * 16 + l16][kk];
        oacc[dt] = wmma_f32(a, bb, oacc[dt]);
      }
    }
    __syncthreads();
  }

  // Epilogue: normalize and write to ao[b][c][n] with c = h*64 + d
  float inv[8];
#pragma unroll
  for (int vv = 0; vv < 8; ++vv) inv[vv] = 1.f / rs[vv];
  const int qr0 = q0 + wid * 16 + 8 * half;
#pragma unroll
  for (int dt = 0; dt < 4; ++dt) {
    const int c = h * HD_ + dt * 16 + l16;
    float* p = ao + ((size_t)b * C_ + c) * N_ + qr0;
    float4 lo, hi;
    lo.x = oacc[dt][0] * inv[0]; lo.y = oacc[dt][1] * inv[1];
    lo.z = oacc[dt][2] * inv[2]; lo.w = oacc[dt][3] * inv[3];
    hi.x = oacc[dt][4] * inv[4]; hi.y = oacc[dt][5] * inv[5];
    hi.z = oacc[dt][6] * inv[6]; hi.w = oacc[dt][7] * inv[7];
    *(float4*)p = lo;
    *(float4*)(p + 4) = hi;
  }
}

// ---------------------------------------------------------------------------
// Kernel 4: output projection + bias + residual.
// out[b,o,n] = sum_c w_proj[o,c]*ao[b,c,n] + b_proj[o] + x[b,o,n]
// ---------------------------------------------------------------------------
__global__ __launch_bounds__(128) void proj_kernel(
    const float* __restrict__ ao, const float* __restrict__ w,
    const float* __restrict__ bias, const float* __restrict__ x,
    float* __restrict__ out) {
  __shared__ float Wl[64][68];  // [o][c]
  __shared__ float Xt[64][68];  // [n][c] (transposed)
  const int n0 = blockIdx.x * 64;
  const int o0 = blockIdx.y * 64;
  const int b  = blockIdx.z;
  const int tid = threadIdx.x;
  const int wid = tid >> 5, lane = tid & 31, l16 = lane & 15, half = lane >> 4;
  const int orow = o0 + wid * 16;
  const float* ab = ao + (size_t)b * C_ * N_;

  v8f acc[4] = {};
  for (int kb = 0; kb < C_; kb += 64) {
    for (int i = tid * 4; i < 64 * 64; i += 128 * 4) {
      int r = i >> 6, cc = i & 63;
      *(float4*)&Wl[r][cc] = *(const float4*)(w + (size_t)(o0 + r) * C_ + kb + cc);
      float4 xv = *(const float4*)(ab + (size_t)(kb + r) * N_ + n0 + cc);
      Xt[cc + 0][r] = xv.x;
      Xt[cc + 1][r] = xv.y;
      Xt[cc + 2][r] = xv.z;
      Xt[cc + 3][r] = xv.w;
    }
    __syncthreads();
#pragma unroll
    for (int ks = 0; ks < 16; ++ks) {
      const int kd = 4 * ks + 2 * half;
      const v2f a = *(const v2f*)&Wl[wid * 16 + l16][kd];
#pragma unroll
      for (int nt = 0; nt < 4; ++nt) {
        const v2f bb = *(const v2f*)&Xt[nt * 16 + l16][kd];
        acc[nt] = wmma_f32(a, bb, acc[nt]);
      }
    }
    __syncthreads();
  }

  float bv[8];
#pragma unroll
  for (int vv = 0; vv < 8; ++vv) bv[vv] = bias[orow + 8 * half + vv];
#pragma unroll
  for (int nt = 0; nt < 4; ++nt) {
    const int n = n0 + nt * 16 + l16;
#pragma unroll
    for (int vv = 0; vv < 8; ++vv) {
      const int o = orow + 8 * half + vv;
      const size_t idx = ((size_t)b * C_ + o) * N_ + n;
      out[idx] = acc[nt][vv] + bv[vv] + x[idx];
    }
  }
}

// ---------------------------------------------------------------------------
extern "C" void kernel_launch(void* const* d_in, const int* in_sizes, int n_in,
                              void* d_out, int out_size, void* d_ws,
                              size_t ws_size, hipStream_t stream) {
  const float* x      = (const float*)d_in[0];
  const float* gamma  = (const float*)d_in[1];
  const float* beta   = (const float*)d_in[2];
  const float* w_qkv  = (const float*)d_in[3];
  const float* b_qkv  = (const float*)d_in[4];
  const float* w_proj = (const float*)d_in[5];
  const float* b_proj = (const float*)d_in[6];
  float* out = (float*)d_out;

  const size_t SZ = (size_t)B_ * C_ * N_;  // 2,097,152 floats
  float* xn = (float*)d_ws;
  float* q  = xn + SZ;
  float* k  = q + SZ;
  float* v  = k + SZ;
  float* ao = v + SZ;  // total 5*SZ floats = ~42 MB

  gn_kernel<<<B_ * GROUPS_, 256, 0, stream>>>(x, gamma, beta, xn);
  qkv_kernel<<<dim3(N_ / 64, (3 * C_) / 64, B_), 128, 0, stream>>>(
      xn, w_qkv, b_qkv, q, k, v);
  attn_kernel<<<dim3(N_ / 64, HEADS_, B_), 128, 0, stream>>>(q, k, v, ao);
  proj_kernel<<<dim3(N_ / 64, C_ / 64, B_), 128, 0, stream>>>(
      ao, w_proj, b_proj, x, out);
}